// QuantumLayer_28595892256866
// MI455X (gfx1250) — compile-verified
//
#include <hip/hip_runtime.h>
#include <hip/hip_bf16.h>
#include <math.h>

typedef __attribute__((ext_vector_type(2))) float v2f;
typedef __attribute__((ext_vector_type(8))) float v8f;

// ---------------------------------------------------------------------------
// Prep kernel (1 thread, trivial cost): build the fixed 16x16 unitary U from
// qweights, reduce it to 4 real symmetric observables M_w = Re(U^H Z_w U),
// and lay out W_pre as a zero-padded 64x16 B-matrix for WMMA.
// ws layout (floats): [0,1024) = M[4][16][16], [1024,2048) = Bpad[64][16]
// ---------------------------------------------------------------------------
__global__ void qlayer_prep(const float* __restrict__ W,   // [4][64]
                            const float* __restrict__ qw,  // [3][4][2]
                            float* __restrict__ Mout,      // [4][16][16]
                            float* __restrict__ Bpad) {    // [64][16]
    if (threadIdx.x != 0 || blockIdx.x != 0) return;

    float Ur[16][16], Ui[16][16];   // column j = U |j>
    for (int k = 0; k < 16; ++k)
        for (int j = 0; j < 16; ++j) { Ur[k][j] = (k == j) ? 1.f : 0.f; Ui[k][j] = 0.f; }

    for (int l = 0; l < 3; ++l) {
        for (int w = 0; w < 4; ++w) {
            const int bit = 1 << (3 - w);           // wire 0 = MSB
            const float ty = qw[(l * 4 + w) * 2 + 0];
            const float tz = qw[(l * 4 + w) * 2 + 1];
            // RY(ty) on wire w
            const float ch = cosf(0.5f * ty), sh = sinf(0.5f * ty);
            for (int k0 = 0; k0 < 16; ++k0) {
                if (k0 & bit) continue;
                const int k1 = k0 | bit;
                for (int j = 0; j < 16; ++j) {
                    float r0 = Ur[k0][j], i0 = Ui[k0][j];
                    float r1 = Ur[k1][j], i1 = Ui[k1][j];
                    Ur[k0][j] = ch * r0 - sh * r1;  Ui[k0][j] = ch * i0 - sh * i1;
                    Ur[k1][j] = sh * r0 + ch * r1;  Ui[k1][j] = sh * i0 + ch * i1;
                }
            }
            // RZ(tz) on wire w: diag(e^{-i tz/2}, e^{+i tz/2})
            const float cz = cosf(0.5f * tz), sz = sinf(0.5f * tz);
            for (int k = 0; k < 16; ++k) {
                for (int j = 0; j < 16; ++j) {
                    float r = Ur[k][j], im = Ui[k][j];
                    if (k & bit) { Ur[k][j] = r * cz - im * sz;  Ui[k][j] = im * cz + r * sz; }
                    else         { Ur[k][j] = r * cz + im * sz;  Ui[k][j] = im * cz - r * sz; }
                }
            }
        }
        // CNOT ring: (0->1),(1->2),(2->3),(3->0)
        for (int w = 0; w < 4; ++w) {
            const int cb = 1 << (3 - w);
            const int tb = 1 << (3 - ((w + 1) & 3));
            for (int k = 0; k < 16; ++k) {
                if ((k & cb) && !(k & tb)) {
                    const int k2 = k | tb;
                    for (int j = 0; j < 16; ++j) {
                        float tr = Ur[k][j]; Ur[k][j] = Ur[k2][j]; Ur[k2][j] = tr;
                        float ti = Ui[k][j]; Ui[k][j] = Ui[k2][j]; Ui[k2][j] = ti;
                    }
                }
            }
        }
    }

    // M_w[i][j] = sum_k z_w(k) * (Ur[k][i]Ur[k][j] + Ui[k][i]Ui[k][j])
    for (int w = 0; w < 4; ++w) {
        const int zb = 1 << (3 - w);
        for (int i = 0; i < 16; ++i)
            for (int j = 0; j < 16; ++j) {
                float acc = 0.f;
                for (int k = 0; k < 16; ++k) {
                    const float t = Ur[k][i] * Ur[k][j] + Ui[k][i] * Ui[k][j];
                    acc += (k & zb) ? -t : t;
                }
                Mout[(w * 16 + i) * 16 + j] = acc;
            }
    }

    // Bpad[k][n] = W_pre[n][k] for n<4, else 0  (B operand, 64x16)
    for (int k = 0; k < 64; ++k)
        for (int n = 0; n < 16; ++n)
            Bpad[k * 16 + n] = (n < 4) ? W[n * 64 + k] : 0.f;
}

// ---------------------------------------------------------------------------
// Main kernel: one block = 256 threads (8 waves) = 256 batch rows.
// Phase 1 (per wave): hoist B (W_pre) into 16 v2f registers; preload both
//   16-row A tiles with one big global clause; run 2 x 16 back-to-back
//   v_wmma_f32_16x16x4_f32 chains (K=64); tanh+bias -> angles to LDS.
// Phase 2 (per lane): own one row; build psi0 product state; 4 quadratic
//   forms against M_w broadcast from LDS; coalesced float4 store.
// ---------------------------------------------------------------------------
__global__ __launch_bounds__(256) void qlayer_main(
    const float* __restrict__ x,     // [B][64]
    const float* __restrict__ bpre,  // [4]
    const float* __restrict__ M,     // [4][16][16]
    const float* __restrict__ Bpad,  // [64][16]
    float* __restrict__ out,         // [B][4]
    int B) {
    __shared__ __align__(16) float sM[1024];
    __shared__ __align__(16) float sBp[1024];
    __shared__ __align__(16) float sAng[256 * 4];

    const int tid = threadIdx.x;
    for (int i = tid; i < 1024; i += 256) { sM[i] = M[i]; sBp[i] = Bpad[i]; }

    const int lane = tid & 31;
    const int wave = tid >> 5;
    const int half = lane >> 4;     // 0: lanes 0-15, 1: lanes 16-31
    const int n16  = lane & 15;
    const long blockRow = (long)blockIdx.x * 256;

    // Uniform, non-divergent bias read (cached); only lanes n16<4 use it.
    const float bb = bpre[n16 & 3];

    __syncthreads();

    // ---- hoist loop-invariant B operand into registers (32 VGPRs) ----
    v2f breg[16];
#pragma unroll
    for (int c = 0; c < 16; ++c) {
        const int k0 = 4 * c;
        breg[c].x = sBp[(k0 + 2 * half) * 16 + n16];      // B rows k0+2h, k0+2h+1
        breg[c].y = sBp[(k0 + 2 * half + 1) * 16 + n16];
    }

    // ---- preload both A tiles (64 VGPRs) ----
    long r0 = blockRow + wave * 32 + n16;        // tile 0 rows
    long r1 = r0 + 16;                           // tile 1 rows
    if (r0 >= B) r0 = B - 1;                     // clamp; EXEC stays all-ones
    if (r1 >= B) r1 = B - 1;
    const float* xr0 = x + r0 * 64 + 2 * half;   // K = k0 + 2*half + {0,1}
    const float* xr1 = x + r1 * 64 + 2 * half;

    v2f a0[16], a1[16];
#pragma unroll
    for (int c = 0; c < 16; ++c) a0[c] = *(const v2f*)(xr0 + 4 * c);
#pragma unroll
    for (int c = 0; c < 16; ++c) a1[c] = *(const v2f*)(xr1 + 4 * c);

    // ---- two back-to-back WMMA chains (K accumulation) ----
    v8f acc0 = {0.f, 0.f, 0.f, 0.f, 0.f, 0.f, 0.f, 0.f};
    v8f acc1 = {0.f, 0.f, 0.f, 0.f, 0.f, 0.f, 0.f, 0.f};
#pragma unroll
    for (int c = 0; c < 16; ++c)
        acc0 = __builtin_amdgcn_wmma_f32_16x16x4_f32(
            false, a0[c], false, breg[c], (short)0, acc0, false, false);
#pragma unroll
    for (int c = 0; c < 16; ++c)
        acc1 = __builtin_amdgcn_wmma_f32_16x16x4_f32(
            false, a1[c], false, breg[c], (short)0, acc1, false, false);

    // C layout: vgpr v, lane l -> row (v + 8*half), col (l & 15)
    if (n16 < 4) {
#pragma unroll
        for (int v = 0; v < 8; ++v) {
            const int rowInTile = v + 8 * half;
            sAng[(wave * 32 + rowInTile) * 4 + n16]      = tanhf(acc0[v] + bb);
            sAng[(wave * 32 + 16 + rowInTile) * 4 + n16] = tanhf(acc1[v] + bb);
        }
    }
    __syncthreads();

    // ---- Phase 2: one row per lane ----
    const long row = blockRow + tid;
    const float4 th = *reinterpret_cast<const float4*>(&sAng[tid * 4]);

    float cw[4], sw[4];
    {
        const float h0 = 0.5f * th.x, h1 = 0.5f * th.y, h2 = 0.5f * th.z, h3 = 0.5f * th.w;
        cw[0] = cosf(h0); sw[0] = sinf(h0);
        cw[1] = cosf(h1); sw[1] = sinf(h1);
        cw[2] = cosf(h2); sw[2] = sinf(h2);
        cw[3] = cosf(h3); sw[3] = sinf(h3);
    }
    // psi0 = kron over wires; index bit (3-w) selects sin for wire w
    float hi[4], lo[4], p[16];
    hi[0] = cw[0] * cw[1]; hi[1] = cw[0] * sw[1]; hi[2] = sw[0] * cw[1]; hi[3] = sw[0] * sw[1];
    lo[0] = cw[2] * cw[3]; lo[1] = cw[2] * sw[3]; lo[2] = sw[2] * cw[3]; lo[3] = sw[2] * sw[3];
#pragma unroll
    for (int i = 0; i < 16; ++i) p[i] = hi[i >> 2] * lo[i & 3];

    float res[4];
#pragma unroll
    for (int w = 0; w < 4; ++w) {
        const float* Mw = &sM[w * 256];
        float acc = 0.f;
#pragma unroll
        for (int i = 0; i < 16; ++i) {
            const float4 m0 = *(const float4*)&Mw[i * 16 + 0];   // ds_load_b128 broadcast
            const float4 m1 = *(const float4*)&Mw[i * 16 + 4];
            const float4 m2 = *(const float4*)&Mw[i * 16 + 8];
            const float4 m3 = *(const float4*)&Mw[i * 16 + 12];
            float t = m0.x * p[0]  + m0.y * p[1]  + m0.z * p[2]  + m0.w * p[3]
                    + m1.x * p[4]  + m1.y * p[5]  + m1.z * p[6]  + m1.w * p[7]
                    + m2.x * p[8]  + m2.y * p[9]  + m2.z * p[10] + m2.w * p[11]
                    + m3.x * p[12] + m3.y * p[13] + m3.z * p[14] + m3.w * p[15];
            acc = fmaf(p[i], t, acc);
        }
        res[w] = acc;
    }

    if (row < B) {
        float4 o; o.x = res[0]; o.y = res[1]; o.z = res[2]; o.w = res[3];
        *reinterpret_cast<float4*>(out + row * 4) = o;            // coalesced b128 store
    }
}

// ---------------------------------------------------------------------------
extern "C" void kernel_launch(void* const* d_in, const int* in_sizes, int n_in,
                              void* d_out, int out_size, void* d_ws, size_t ws_size,
                              hipStream_t stream) {
    const float* x    = (const float*)d_in[0];   // [B][64]
    const float* Wpre = (const float*)d_in[1];   // [4][64]
    const float* bpre = (const float*)d_in[2];   // [4]
    const float* qw   = (const float*)d_in[3];   // [3][4][2]
    float* out = (float*)d_out;

    float* wsf  = (float*)d_ws;
    float* M    = wsf;          // 1024 floats
    float* Bpad = wsf + 1024;   // 1024 floats

    const int B = in_sizes[0] / 64;

    qlayer_prep<<<1, 1, 0, stream>>>(Wpre, qw, M, Bpad);

    const int nblk = (B + 255) / 256;
    qlayer_main<<<nblk, 256, 0, stream>>>(x, bpre, M, Bpad, out, B);
}